// KANClassification_67946382623146
// MI455X (gfx1250) — compile-verified
//
#include <hip/hip_runtime.h>
#include <hip/hip_bf16.h>

typedef __attribute__((ext_vector_type(2))) float v2f;
typedef __attribute__((ext_vector_type(8))) float v8f;

// ---------------------------------------------------------------------------
// Cubic B-spline basis (Cox-de Boor), N_BASIS=7, knots:
// [-1,-1,-1, -1,-0.5,0,0.5,1, 1,1,1]
// ---------------------------------------------------------------------------
__device__ __forceinline__ void bspline7(float x, float* __restrict__ out) {
    const float t[11] = {-1.f,-1.f,-1.f,-1.f,-0.5f,0.f,0.5f,1.f,1.f,1.f,1.f};
    float xe = fminf(fmaxf(x, -1.0f), 1.0f - 1e-6f);
    float N[10];
#pragma unroll
    for (int j = 0; j < 10; ++j)
        N[j] = (xe >= t[j] && xe < t[j + 1]) ? 1.0f : 0.0f;
#pragma unroll
    for (int d = 1; d <= 3; ++d) {
#pragma unroll
        for (int j = 0; j < 10 - 3; ++j) {   // only need first 10-d; 7 is enough for all d
            if (j < 10 - d) {
                float ld = t[j + d] - t[j];
                float rd = t[j + d + 1] - t[j + 1];
                float left  = (ld > 0.f) ? (xe - t[j]) / ld * N[j] : 0.0f;
                float right = (rd > 0.f) ? (t[j + d + 1] - xe) / rd * N[j + 1] : 0.0f;
                N[j] = left + right;
            }
        }
        // elements beyond 10-d are stale but never read at final level (<7)
        // except intermediate levels need up to 10-d entries; handle exactly:
        if (d == 1) { int j = 7; { float ld=t[j+1]-t[j], rd=t[j+2]-t[j+1];
            float l2=(ld>0.f)?(xe-t[j])/ld*N[j]:0.f, r2=(rd>0.f)?(t[j+2]-xe)/rd*N[j+1]:0.f; N[j]=l2+r2; }
            j = 8; { float ld=t[j+1]-t[j], rd=t[j+2]-t[j+1];
            float l2=(ld>0.f)?(xe-t[j])/ld*N[j]:0.f, r2=(rd>0.f)?(t[j+2]-xe)/rd*N[j+1]:0.f; N[j]=l2+r2; } }
        if (d == 2) { int j = 7; { float ld=t[j+2]-t[j], rd=t[j+3]-t[j+1];
            float l2=(ld>0.f)?(xe-t[j])/ld*N[j]:0.f, r2=(rd>0.f)?(t[j+3]-xe)/rd*N[j+1]:0.f; N[j]=l2+r2; } }
    }
#pragma unroll
    for (int j = 0; j < 7; ++j) out[j] = N[j];
}

// ---------------------------------------------------------------------------
// KAN conv as WMMA GEMM: out[n, p] = sum_k bases[p, k] * Wc[n, k]
//   in : [CIN][Hin][Hin], Wc : [COUT][E*7], out : [COUT][Hout*Hout]
//   E = CIN*9, K = E*7 padded to KPAD (multiple of 4)
// Each wave handles a tile of 16 pixels; f32 16x16x4 WMMA, KPAD/4 steps.
// ---------------------------------------------------------------------------
template <int CIN, int COUT, int E, int KPAD, int WPB>
__global__ void kan_conv_wmma(const float* __restrict__ in,
                              const float* __restrict__ Wc,
                              float* __restrict__ out,
                              int Hin, int Hout) {
    constexpr int K7 = E * 7;
    __shared__ float Alds[WPB][16][KPAD];
    __shared__ float Blds[KPAD][16];

    const int tid  = threadIdx.x;
    const int wave = tid >> 5;
    const int lane = tid & 31;
    const int Npix = Hout * Hout;

    // Stage B matrix (K x 16), channels >= COUT padded with zeros.
    for (int idx = tid; idx < KPAD * 16; idx += blockDim.x) {
        int k = idx >> 4, n = idx & 15;
        Blds[k][n] = (n < COUT && k < K7) ? Wc[n * K7 + k] : 0.0f;
    }

    // Compute A fragments (basis values) for this wave's 16 pixels.
    const int tile    = blockIdx.x * WPB + wave;
    const int pixBase = tile * 16;
    const int m       = lane & 15;
    const int pix     = pixBase + m;
    {
        const int eHalf = (E + 1) / 2;
        const int e0 = (lane < 16) ? 0 : eHalf;
        const int e1 = (lane < 16) ? eHalf : E;
        const bool valid = (pix < Npix);
        const int pi = valid ? (pix / Hout) : 0;
        const int pj = valid ? (pix - pi * Hout) : 0;
        for (int e = e0; e < e1; ++e) {
            int c  = e / 9, r = e - 9 * c;
            int di = r / 3, dj = r - 3 * di;
            float xv = valid ? in[(c * Hin + (pi + di)) * Hin + (pj + dj)] : 0.0f;
            float bs[7];
            bspline7(xv, bs);
#pragma unroll
            for (int k = 0; k < 7; ++k)
                Alds[wave][m][e * 7 + k] = valid ? bs[k] : 0.0f;
        }
        if (lane < 16) {
            for (int k = K7; k < KPAD; ++k) Alds[wave][m][k] = 0.0f;
        }
    }
    __syncthreads();

    // WMMA accumulation over K.
    v8f acc = {};
    const int khalf = (lane >> 4) ? 2 : 0;
    const int n     = lane & 15;
#pragma unroll 4
    for (int k0 = 0; k0 < KPAD; k0 += 4) {
        v2f a, b;
        a.x = Alds[wave][m][k0 + khalf];
        a.y = Alds[wave][m][k0 + khalf + 1];
        b.x = Blds[k0 + khalf][n];
        b.y = Blds[k0 + khalf + 1][n];
        acc = __builtin_amdgcn_wmma_f32_16x16x4_f32(
            false, a, false, b, (short)0, acc, false, false);
    }

    // Store D: lane owns column n; VGPR r -> row M = r + 8*(lane>=16).
    if (n < COUT) {
        const int mAdd = (lane >> 4) ? 8 : 0;
#pragma unroll
        for (int r = 0; r < 8; ++r) {
            int p = pixBase + r + mAdd;
            if (p < Npix) out[n * Npix + p] = acc[r];
        }
    }
}

// ---------------------------------------------------------------------------
// Basis expansion for the dense layer: Bf[i*7+k] = B_k(v[i])
// ---------------------------------------------------------------------------
__global__ void kan_basis_kernel(const float* __restrict__ v,
                                 float* __restrict__ Bf, int n) {
    int i = blockIdx.x * blockDim.x + threadIdx.x;
    if (i >= n) return;
    float bs[7];
    bspline7(v[i], bs);
#pragma unroll
    for (int k = 0; k < 7; ++k) Bf[i * 7 + k] = bs[k];
}

// ---------------------------------------------------------------------------
// Memory-bound dense: partial dot products of 32 rows of W3 (flat e*7+k)
// against Bf. float4 streaming loads; deterministic tree reduction.
// ---------------------------------------------------------------------------
__global__ void kan_dense_dot(const float* __restrict__ W3,
                              const float* __restrict__ Bf,
                              float* __restrict__ partials,
                              int n4, int bpr) {
    __shared__ float red[256];
    const int o  = blockIdx.x / bpr;
    const int ch = blockIdx.x - o * bpr;
    const float4* __restrict__ w  = (const float4*)W3 + (size_t)o * n4;
    const float4* __restrict__ bv = (const float4*)Bf;
    float acc = 0.0f;
    for (int i = ch * (int)blockDim.x + (int)threadIdx.x; i < n4;
         i += bpr * (int)blockDim.x) {
        float4 a = w[i];
        float4 b = bv[i];
        acc = fmaf(a.x, b.x, acc);
        acc = fmaf(a.y, b.y, acc);
        acc = fmaf(a.z, b.z, acc);
        acc = fmaf(a.w, b.w, acc);
    }
    red[threadIdx.x] = acc;
    __syncthreads();
    for (int s = blockDim.x / 2; s > 0; s >>= 1) {
        if ((int)threadIdx.x < s) red[threadIdx.x] += red[threadIdx.x + s];
        __syncthreads();
    }
    if (threadIdx.x == 0) partials[blockIdx.x] = red[0];
}

// ---------------------------------------------------------------------------
// Finish: reduce partials -> h1[32]; basis(h1); out[10] = W4 . B2
// Single block (one wave, 32 threads) for deterministic order.
// ---------------------------------------------------------------------------
__global__ void kan_finish(const float* __restrict__ partials,
                           const float* __restrict__ W4,
                           float* __restrict__ out, int bpr) {
    __shared__ float h1s[32];
    __shared__ float B2[32][7];
    const int t = threadIdx.x;
    {
        float s = 0.0f;
        for (int c = 0; c < bpr; ++c) s += partials[t * bpr + c];
        h1s[t] = s;
    }
    __syncthreads();
    {
        float bs[7];
        bspline7(h1s[t], bs);
#pragma unroll
        for (int k = 0; k < 7; ++k) B2[t][k] = bs[k];
    }
    __syncthreads();
    if (t < 10) {
        float s = 0.0f;
        for (int e = 0; e < 32; ++e)
#pragma unroll
            for (int k = 0; k < 7; ++k)
                s = fmaf(W4[(t * 32 + e) * 7 + k], B2[e][k], s);
        out[t] = s;
    }
}

// ---------------------------------------------------------------------------
extern "C" void kernel_launch(void* const* d_in, const int* in_sizes, int n_in,
                              void* d_out, int out_size, void* d_ws, size_t ws_size,
                              hipStream_t stream) {
    const float* x  = (const float*)d_in[0];  // [1,3,384,384]
    const float* W1 = (const float*)d_in[1];  // [2,27,7]
    const float* W2 = (const float*)d_in[2];  // [1,18,7]
    const float* W3 = (const float*)d_in[3];  // [32,144400,7]
    const float* W4 = (const float*)d_in[4];  // [10,32,7]
    float* out = (float*)d_out;               // [10]

    constexpr int H0 = 384, H1 = 382, H2 = 380;
    constexpr int Npix1 = H1 * H1;            // 145924
    constexpr int Npix2 = H2 * H2;            // 144400
    constexpr int L     = Npix2 * 7;          // 1010800
    constexpr int BPR   = 240;                // blocks per output row (dense)

    float* ws = (float*)d_ws;
    float* h  = ws;                           // [2, 382, 382]
    float* v  = h + 2 * Npix1;                // [144400]
    float* Bf = v + Npix2;                    // [144400*7]
    float* pp = Bf + L;                       // [32*BPR]

    // conv1: CIN=3, COUT=2, E=27, K=189 -> KPAD=192
    {
        constexpr int WPB = 4;
        int tiles  = (Npix1 + 15) / 16;       // 9121
        int blocks = (tiles + WPB - 1) / WPB;
        kan_conv_wmma<3, 2, 27, 192, WPB>
            <<<blocks, WPB * 32, 0, stream>>>(x, W1, h, H0, H1);
    }
    // conv2: CIN=2, COUT=1, E=18, K=126 -> KPAD=128
    {
        constexpr int WPB = 4;
        int tiles  = (Npix2 + 15) / 16;       // 9025
        int blocks = (tiles + WPB - 1) / WPB;
        kan_conv_wmma<2, 1, 18, 128, WPB>
            <<<blocks, WPB * 32, 0, stream>>>(h, W2, v, H1, H2);
    }
    // basis expansion for dense layer
    kan_basis_kernel<<<(Npix2 + 255) / 256, 256, 0, stream>>>(v, Bf, Npix2);

    // big memory-bound dot: 32 rows x 129MB total of W3
    kan_dense_dot<<<32 * BPR, 256, 0, stream>>>(W3, Bf, pp, L / 4, BPR);

    // final reduction + dense2
    kan_finish<<<1, 32, 0, stream>>>(pp, W4, out, BPR);

    (void)in_sizes; (void)n_in; (void)out_size; (void)ws_size;
}